// SimpleT5Model_52862457479597
// MI455X (gfx1250) — compile-verified
//
#include <hip/hip_runtime.h>
#include <hip/hip_bf16.h>

// Shapes fixed by the reference: feats (B,T,V,C) f32, seq_len (B,) i32,
// out (B,V,T,C) f32.
namespace {
constexpr int B_ = 8;
constexpr int T_ = 1024;   // == max_len (power of two)
constexpr int V_ = 22;
constexpr int C_ = 512;
constexpr int ROWS_PER_BLOCK = 8;   // one wave32 per (b,v,t) row
constexpr int THREADS = 256;        // 8 waves / block
}

typedef __attribute__((ext_vector_type(4))) float f4;

__global__ __launch_bounds__(THREADS, 8)
void stretch_lerp_kernel(const float* __restrict__ in,
                         const int*   __restrict__ seq_len,
                         float*       __restrict__ out) {
  const int tid  = threadIdx.x;
  const int lane = tid & 31;   // lane within wave -> channel group
  const int wrow = tid >> 5;   // wave index within block -> row

  // Global row index r = (b*V + v)*T + t.  T divisible by ROWS_PER_BLOCK, so
  // all 8 rows of a block share the same (b,v); bv is block-uniform (SALU).
  const int row = blockIdx.x * ROWS_PER_BLOCK + wrow;
  const int t   = row & (T_ - 1);                  // wave-uniform
  const int bv  = (int)(blockIdx.x >> 7);          // (b*V + v), block-uniform
  const int v   = bv % V_;
  const int b   = bv / V_;

  // Wave-uniform scalar math (one issue per wave, not per lane).
  const int      L    = seq_len[b];                 // uniform -> s_load
  const unsigned step = (unsigned)T_ / (unsigned)L; // data-dependent stride
  const unsigned i    = (unsigned)t / step;
  const unsigned j    = (unsigned)t - i * step;
  const float    ratio = (float)j / (float)step;
  const bool     valid = (int)i < (L - 1);          // wave-uniform branch
  const int      ic    = ((int)i < (T_ - 2)) ? (int)i : (T_ - 2);

  // Row bases: input row (b, ic, v, :) and (b, ic+1, v, :); output row (bv, t, :).
  // All byte offsets < 2^31 -> 32-bit voffset + SGPR base (GVS addressing).
  const float* __restrict__ pa = in  + (size_t)((b * T_ + ic) * V_ + v) * C_ + lane * 4;
  float*       __restrict__ po = out + (size_t)row * C_ + lane * 4;

  if (valid) {
    // Warm the source row two time-steps ahead (speculative; OOB is dropped).
    if (ic < T_ - 2) {
      __builtin_prefetch(pa + 2 * (V_ * C_), 0, 1);  // global_prefetch_b8
    }
#pragma unroll
    for (int k = 0; k < 4; ++k) {
      // 32 lanes x float4 = 128 floats per k-chunk; 4 chunks cover C=512.
      f4 a  = *(const f4*)(pa + k * 128);                // global_load_b128
      f4 bb = *(const f4*)(pa + k * 128 + V_ * C_);      // next source row
      f4 r;
      r.x = fmaf(ratio, bb.x - a.x, a.x);
      r.y = fmaf(ratio, bb.y - a.y, a.y);
      r.z = fmaf(ratio, bb.z - a.z, a.z);
      r.w = fmaf(ratio, bb.w - a.w, a.w);
      // Output is write-once: non-temporal store keeps the re-read input hot
      // in L2 and streams the result straight toward HBM.
      __builtin_nontemporal_store(r, (f4*)(po + k * 128));
    }
  } else {
    const f4 z = {0.0f, 0.0f, 0.0f, 0.0f};
#pragma unroll
    for (int k = 0; k < 4; ++k) {
      __builtin_nontemporal_store(z, (f4*)(po + k * 128));
    }
  }
}

extern "C" void kernel_launch(void* const* d_in, const int* in_sizes, int n_in,
                              void* d_out, int out_size, void* d_ws, size_t ws_size,
                              hipStream_t stream) {
  (void)in_sizes; (void)n_in; (void)out_size; (void)d_ws; (void)ws_size;

  const float* feats   = (const float*)d_in[0];  // (B, T, V, C) f32
  const int*   seq_len = (const int*)d_in[1];    // (B,) i32
  float*       out     = (float*)d_out;          // (B, V, T, C) f32

  const int total_rows = B_ * V_ * T_;                    // 180224
  const int blocks     = total_rows / ROWS_PER_BLOCK;     // 22528

  stretch_lerp_kernel<<<dim3(blocks), dim3(THREADS), 0, stream>>>(feats, seq_len, out);
}